// ADMM_Net_30786325577820
// MI455X (gfx1250) — compile-verified
//
#include <hip/hip_runtime.h>
#include <math.h>

typedef __attribute__((ext_vector_type(2))) float v2f;
typedef __attribute__((ext_vector_type(8))) float v8f;
typedef __attribute__((ext_vector_type(4))) unsigned int v4u;
typedef __attribute__((ext_vector_type(8))) int v8i;
typedef __attribute__((ext_vector_type(4))) int v4i;

#define Bsz 8192
#define Nn 128
#define Kk 64
#define Ll 20
#define SW 65   // LDS row stride for H (odd -> conflict-free row access across lanes)
#define SA 65   // LDS row stride for HtH / Cholesky factor

#if defined(__has_builtin)
#if __has_builtin(__builtin_amdgcn_tensor_load_to_lds) && __has_builtin(__builtin_amdgcn_s_wait_tensorcnt)
#define USE_TDM 1
#endif
#endif

__device__ __forceinline__ float demod_nl(float x, float th) {
    const float Sc = 0.31622776601683794f;      // 1/sqrt(10)
    const float m  = 2.0f * Sc;                 // MIDS = {-2S, 0, 2S}
    return Sc * (tanhf(th * (x + m)) + tanhf(th * x) + tanhf(th * (x - m)));
}

__device__ __forceinline__ float demod_proj(float xh) {
    const float Sc = 0.31622776601683794f;
    float u = rintf((xh / Sc + 3.0f) * 0.5f) * 2.0f - 3.0f;
    u = fminf(3.0f, fmaxf(-3.0f, u));
    return u * Sc;
}

__global__ void zero_tail_kernel(float* __restrict__ out) {
    int i = threadIdx.x;
    if (i < 2 * Ll) out[2 * Bsz * Kk + i] = 0.0f;
}

__global__ __launch_bounds__(256) void admm_net_kernel(
    const float* __restrict__ gH,     // (B, N, K)
    const float* __restrict__ gY,     // (B, N, 1)
    const float* __restrict__ gX,     // (B, K, 1)
    const float* __restrict__ gDelta, // (L,1)
    const float* __restrict__ gTheta,
    const float* __restrict__ gAlpha,
    const float* __restrict__ gBeta,
    const float* __restrict__ gGamma, // (L,)
    float* __restrict__ out)
{
    __shared__ float sH[Nn * SW];     // H tile, 128 x 64 (stride 65)
    __shared__ float sM[Kk * SA];     // HtH -> Cholesky factor L
    __shared__ float sY[Nn];
    __shared__ float sXr[Kk];         // reference X (for loss/ser)
    __shared__ float sRhs[Kk];        // H^T y
    __shared__ float sx[Kk];          // iterate x
    __shared__ float sy2[Kk];         // fwd-subst intermediate
    __shared__ float sE[Nn], sVh[Nn], sVp[Nn], sT1[Nn];
    __shared__ float sRedL[Kk], sRedS[Kk];

    const int   b    = blockIdx.x;
    const int   tid  = threadIdx.x;
    const int   lane = tid & 31;
    const int   wave = tid >> 5;

#ifdef USE_TDM
    // ---- H tile via Tensor Data Mover: one DMA descriptor moves the whole
    // 128x64 fp32 tile HBM -> LDS, inserting 1 DWORD of padding every 64
    // DWORDs (pad_interval=5, pad_amount=0) to produce the stride-65
    // bank-conflict-free layout directly. Issued by wave 0 only (EXEC is
    // ignored by TDM; every issuing wave would repeat the DMA).
    if (wave == 0) {
        const unsigned long long gaddr =
            (unsigned long long)(uintptr_t)(gH + (size_t)b * (Nn * Kk));
        const unsigned int ldsa = (unsigned int)(uintptr_t)(&sH[0]); // LDS byte offset
        v4u g0;
        g0[0] = 1u;                          // count=1 (valid), user mode, no gather
        g0[1] = ldsa;                        // lds_addr
        g0[2] = (unsigned int)gaddr;         // global_addr[31:0]
        g0[3] = (unsigned int)((gaddr >> 32) & 0x01FFFFFFu) | (2u << 30); // addr[56:32] | type=2
        v8i g1;
        g1[0] = (2 << 16)      // data_size = 4B
              | (1 << 20)      // pad_enable
              | (5 << 22)      // pad_interval: 64 DWORDs
              | (0 << 25);     // pad_amount: 1 DWORD
        g1[1] = (Kk << 16);    // tensor_dim0 = 64 (bits 79:48 low half)
        g1[2] = (Nn << 16);    // tensor_dim1 = 128 (bits 111:80 low half)
        g1[3] = (Kk << 16);    // tile_dim0 = 64 (bits 127:112)
        g1[4] = Nn;            // tile_dim1 = 128 (bits 143:128)
        g1[5] = Kk;            // tensor_dim0_stride = 64 (bits 207:160 low 32)
        g1[6] = 0;
        g1[7] = 0;
        v4i gz4 = {0, 0, 0, 0};              // 2-D tensor: groups 2/3 unused
        v8i gz8 = {0, 0, 0, 0, 0, 0, 0, 0};  // trailing group (clang-23 6-arg form)
        __builtin_amdgcn_tensor_load_to_lds(g0, g1, gz4, gz4, gz8, 0);
        __builtin_amdgcn_s_wait_tensorcnt(0);
    }
#else
    const float* Hb = gH + (size_t)b * (Nn * Kk);
    for (int idx = tid; idx < Nn * Kk; idx += 256) {
        int n = idx >> 6, k = idx & 63;
        sH[n * SW + k] = Hb[idx];
    }
#endif
    if (tid < Nn) sY[tid]  = gY[(size_t)b * Nn + tid];
    if (tid < Kk) sXr[tid] = gX[(size_t)b * Kk + tid];
    __syncthreads();

    // ---- HtH = H^T * H via V_WMMA_F32_16X16X4_F32 ----
    // 64x64 output = 4x4 grid of 16x16 tiles; each of 8 waves owns 2 tiles
    // sharing the same A (row block of H^T).
    {
        const int ti  = wave >> 1;          // A row-tile (0..3)
        const int tj0 = (wave & 1) * 2;     // first B col-tile (0 or 2)
        const int lm  = lane & 15;
        const int kh  = (lane >> 4) << 1;   // lanes 0-15: K offset 0; 16-31: K offset 2
        const int ca  = ti  * 16 + lm;      // A column in H (== H^T row)
        const int cb0 = tj0 * 16 + lm;
        const int cb1 = cb0 + 16;
        v8f c0 = {}; v8f c1 = {};
        for (int k0 = 0; k0 < Nn; k0 += 4) {
            const float* row0 = &sH[(k0 + kh) * SW];
            const float* row1 = row0 + SW;
            v2f a;  a.x  = row0[ca];  a.y  = row1[ca];
            v2f b0; b0.x = row0[cb0]; b0.y = row1[cb0];
            v2f b1; b1.x = row0[cb1]; b1.y = row1[cb1];
            c0 = __builtin_amdgcn_wmma_f32_16x16x4_f32(false, a, false, b0,
                                                       (short)0, c0, false, false);
            c1 = __builtin_amdgcn_wmma_f32_16x16x4_f32(false, a, false, b1,
                                                       (short)0, c1, false, false);
        }
        const int rbase = ti * 16 + ((lane >> 4) << 3); // lanes16-31 hold M=8..15
        #pragma unroll
        for (int r = 0; r < 8; ++r) {
            sM[(rbase + r) * SA + cb0] = c0[r];
            sM[(rbase + r) * SA + cb1] = c1[r];
        }
    }
    __syncthreads();

    // ---- rhs = H^T y (same-row access across lanes => conflict free) ----
    if (tid < Kk) {
        float s = 0.0f;
        for (int n = 0; n < Nn; ++n) s += sH[n * SW + tid] * sY[n];
        sRhs[tid] = s;
    }
    __syncthreads();

    // ---- Cholesky factorization of HtH (right-looking, in LDS) ----
    for (int j = 0; j < Kk; ++j) {
        if (tid == 0) sM[j * SA + j] = sqrtf(sM[j * SA + j]);
        __syncthreads();
        float dinv = 1.0f / sM[j * SA + j];
        for (int i = j + 1 + tid; i < Kk; i += 256) sM[i * SA + j] *= dinv;
        __syncthreads();
        for (int i = j + 1 + tid; i < Kk; i += 256) {
            float lij = sM[i * SA + j];
            for (int k = j + 1; k <= i; ++k)
                sM[i * SA + k] -= lij * sM[k * SA + j];
        }
        __syncthreads();
    }

    // ---- forward substitution: L y2 = rhs (wave 0, shfl reduction) ----
    for (int i = 0; i < Kk; ++i) {
        if (tid < 32) {
            float s = 0.0f;
            for (int k = lane; k < i; k += 32) s += sM[i * SA + k] * sy2[k];
            for (int off = 16; off > 0; off >>= 1) s += __shfl_xor(s, off, 32);
            if (lane == 0) sy2[i] = (sRhs[i] - s) / sM[i * SA + i];
        }
        __syncthreads();
    }
    // ---- backward substitution: L^T x0 = y2 ----
    for (int i = Kk - 1; i >= 0; --i) {
        if (tid < 32) {
            float s = 0.0f;
            for (int k = i + 1 + lane; k < Kk; k += 32) s += sM[k * SA + i] * sx[k];
            for (int off = 16; off > 0; off >>= 1) s += __shfl_xor(s, off, 32);
            if (lane == 0) sx[i] = (sy2[i] - s) / sM[i * SA + i];
        }
        __syncthreads();
    }

    if (tid < Nn) { sE[tid] = 0.0f; sVh[tid] = 0.0f; sVp[tid] = 0.0f; }
    __syncthreads();

    // ---- 20 fused ADMM layers, all operands resident in LDS ----
    const float Sc    = 0.31622776601683794f;
    const float invBK = 1.0f / (float)(Bsz * Kk);
    for (int l = 0; l < Ll; ++l) {
        const float d  = gDelta[l];
        const float th = gTheta[l];
        const float al = gAlpha[l];
        const float be = gBeta[l];
        const float ga = gGamma[l];

        if (tid < Nn) sT1[tid] = sE[tid] + (1.0f - 2.0f * be) * sVh[tid];
        __syncthreads();

        if (tid < Kk) {
            float s = 0.0f;
            for (int n = 0; n < Nn; ++n) s += sH[n * SW + tid] * sT1[n];
            float xv = demod_nl(sx[tid] - d * s, th);
            sx[tid] = xv;
            float xh  = demod_nl(xv, 100.0f);
            float xr  = demod_proj(xh);
            float dd  = sXr[tid] - xh;
            sRedL[tid] = dd * dd;
            sRedS[tid] = (fabsf(xr - sXr[tid]) > Sc * 0.001f) ? 1.0f : 0.0f;
        }
        __syncthreads();

        if (tid < Nn) {
            float s = 0.0f;
            for (int k = 0; k < Kk; ++k) s += sH[tid * SW + k] * sx[k];
            float e  = s - sY[tid];
            float v  = al * e + (1.0f - al * be) * sVh[tid];
            float vh = v + ga * (v - sVp[tid]);
            sE[tid] = e; sVp[tid] = v; sVh[tid] = vh;
        }
        if (tid == 0) {
            float sl = 0.0f, ss = 0.0f;
            for (int k = 0; k < Kk; ++k) { sl += sRedL[k]; ss += sRedS[k]; }
            atomicAdd(&out[2 * Bsz * Kk + l],      sl * invBK);
            atomicAdd(&out[2 * Bsz * Kk + Ll + l], ss * invBK);
        }
        __syncthreads();
    }

    // ---- final outputs: x_hat, x_result ----
    if (tid < Kk) {
        float xh = demod_nl(sx[tid], 100.0f);
        out[(size_t)b * Kk + tid]                    = xh;
        out[(size_t)Bsz * Kk + (size_t)b * Kk + tid] = demod_proj(xh);
    }
}

extern "C" void kernel_launch(void* const* d_in, const int* in_sizes, int n_in,
                              void* d_out, int out_size, void* d_ws, size_t ws_size,
                              hipStream_t stream) {
    (void)in_sizes; (void)n_in; (void)out_size; (void)d_ws; (void)ws_size;
    const float* H     = (const float*)d_in[0];
    const float* Y     = (const float*)d_in[1];
    // d_in[2] = H_T : intentionally unused (we read H once; halves HBM traffic)
    const float* X     = (const float*)d_in[3];
    const float* delta = (const float*)d_in[4];
    const float* theta = (const float*)d_in[5];
    const float* alpha = (const float*)d_in[6];
    const float* beta  = (const float*)d_in[7];
    const float* gamma = (const float*)d_in[8];
    float* out = (float*)d_out;

    hipLaunchKernelGGL(zero_tail_kernel, dim3(1), dim3(64), 0, stream, out);
    hipLaunchKernelGGL(admm_net_kernel, dim3(Bsz), dim3(256), 0, stream,
                       H, Y, X, delta, theta, alpha, beta, gamma, out);
}